// VoxelizationLayer_3582002725062
// MI455X (gfx1250) — compile-verified
//
#include <hip/hip_runtime.h>

// -------------------- problem constants (match reference) --------------------
#define GXD 512
#define GYD 512
#define PMAX 32          // max points per pillar
#define CH 4             // channels per point
#define SCAN_BLK 1024    // 32 wave32 waves per scan block

typedef float v4f __attribute__((ext_vector_type(4)));

// -------------------- helpers --------------------
__device__ __forceinline__ int pillar_flat(const float* __restrict__ pts,
                                           const int* __restrict__ bidx, int i) {
    // reference: xy = int32(p[:, :2] / 0.2f) (truncate toward zero), clip [0,511]
    float px = pts[(size_t)i * CH + 0];
    float py = pts[(size_t)i * CH + 1];
    int x = (int)(px / 0.2f);
    int y = (int)(py / 0.2f);
    x = min(max(x, 0), GXD - 1);
    y = min(max(y, 0), GYD - 1);
    return bidx[i] * (GXD * GYD) + x * GYD + y;
}

// -------------------- kernels --------------------
__global__ void k_init(int* __restrict__ cnt, int* __restrict__ last_pos, int G) {
    int g = blockIdx.x * blockDim.x + threadIdx.x;
    if (g < G) { cnt[g] = 0; last_pos[g] = -1; }
}

__global__ void k_count(const float* __restrict__ pts, const int* __restrict__ bidx,
                        int* __restrict__ cnt, int N) {
    int i = blockIdx.x * blockDim.x + threadIdx.x;
    if (i >= N) return;
    atomicAdd(&cnt[pillar_flat(pts, bidx, i)], 1);
}

// Per-block inclusive scan of occupancy flags; pack (scan<<2 | maskBit | occBit).
__global__ void k_blockscan(const int* __restrict__ cnt, int* __restrict__ rnk,
                            int* __restrict__ part) {
    __shared__ int s[SCAN_BLK];
    int tid = threadIdx.x;
    int g = blockIdx.x * SCAN_BLK + tid;
    int c = cnt[g];
    int occ = (c > 0) ? 1 : 0;
    s[tid] = occ;
    __syncthreads();
#pragma unroll
    for (int off = 1; off < SCAN_BLK; off <<= 1) {
        int v = (tid >= off) ? s[tid - off] : 0;
        __syncthreads();
        s[tid] += v;
        __syncthreads();
    }
    int flags = occ | ((c <= PMAX) ? 2 : 0);
    rnk[g] = (s[tid] << 2) | flags;           // block-local inclusive scan + flags
    if (tid == SCAN_BLK - 1) part[blockIdx.x] = s[tid];
}

// Single-block exclusive scan of block partial sums (NB <= 1024).
__global__ void k_scanpart(int* __restrict__ part, int NB) {
    __shared__ int s[SCAN_BLK];
    int tid = threadIdx.x;
    int v = (tid < NB) ? part[tid] : 0;
    s[tid] = v;
    __syncthreads();
#pragma unroll
    for (int off = 1; off < SCAN_BLK; off <<= 1) {
        int t = (tid >= off) ? s[tid - off] : 0;
        __syncthreads();
        s[tid] += t;
        __syncthreads();
    }
    if (tid < NB) part[tid] = s[tid] - v;     // exclusive
}

// rnk[g] = rank (cumsum(occ)-1) if occupied && count<=PMAX, else -1.
__global__ void k_finalize(int* __restrict__ rnk, const int* __restrict__ part, int G) {
    int g = blockIdx.x * blockDim.x + threadIdx.x;
    if (g >= G) return;
    int v = rnk[g];
    int rank = (v >> 2) + part[g >> 10] - 1;  // SCAN_BLK == 1024
    rnk[g] = ((v & 3) == 3) ? rank : -1;
}

// last-write-wins: max point index among masked points of each rank.
__global__ void k_winner(const float* __restrict__ pts, const int* __restrict__ bidx,
                         const int* __restrict__ rnk, int* __restrict__ last_pos, int N) {
    int i = blockIdx.x * blockDim.x + threadIdx.x;
    if (i >= N) return;
    int r = rnk[pillar_flat(pts, bidx, i)];
    if (r >= 0) atomicMax(&last_pos[r], i);
}

// One lane per (pillar-slot g, point-slot): one NT b128 store, fully coalesced.
__global__ void k_write(const float* __restrict__ pts, const int* __restrict__ last_pos,
                        float* __restrict__ out, int G) {
    int t = blockIdx.x * blockDim.x + threadIdx.x;  // up to G*32 = 16.7M, fits int
    int g = t >> 5;
    if (g >= G) return;
    int slot = t & 31;
    int lp = last_pos[g];                           // uniform across the 32-lane group
    v4f v = (v4f)(0.0f);
    if (lp >= 0) v = *(const v4f*)(pts + (size_t)lp * CH);
    float* dst = out + (size_t)g * (PMAX * CH) + slot * CH;
    __builtin_nontemporal_store(v, (v4f*)dst);      // 268MB stream: bypass-cache hint
}

// voxel_counts tail: reference never writes it -> zeros (after k_write consumed it).
__global__ void k_zero_tail(int* __restrict__ vc, int G) {
    int g = blockIdx.x * blockDim.x + threadIdx.x;
    if (g < G) vc[g] = 0;
}

// -------------------- launch --------------------
extern "C" void kernel_launch(void* const* d_in, const int* in_sizes, int n_in,
                              void* d_out, int out_size, void* d_ws, size_t ws_size,
                              hipStream_t stream) {
    const float* pts = (const float*)d_in[0];
    const int*   bidx = (const int*)d_in[1];
    // d_in[2] (num_point_clouds) is a device scalar; derive B on host instead:
    // out_size = B * GX*GY * (PMAX*CH + 1)
    const int N = in_sizes[0] / CH;
    const int B = out_size / (GXD * GYD * (PMAX * CH + 1));
    const int G = B * GXD * GYD;                    // total pillar slots (multiple of 1024)
    const int NB = G / SCAN_BLK;                    // <= 1024 for B <= 4

    int* cnt  = (int*)d_ws;                         // G ints
    int* rnk  = cnt + G;                            // G ints
    int* part = rnk + G;                            // NB ints  (total ws ~4.2 MB)

    float* out = (float*)d_out;
    // reuse the voxel_counts tail of d_out as last_pos scratch; re-zeroed at the end
    int* last_pos = (int*)(out + (size_t)G * (PMAX * CH));

    const int TB = 256;
    k_init     <<<(G + TB - 1) / TB, TB, 0, stream>>>(cnt, last_pos, G);
    k_count    <<<(N + TB - 1) / TB, TB, 0, stream>>>(pts, bidx, cnt, N);
    k_blockscan<<<NB, SCAN_BLK, 0, stream>>>(cnt, rnk, part);
    k_scanpart <<<1, SCAN_BLK, 0, stream>>>(part, NB);
    k_finalize <<<(G + TB - 1) / TB, TB, 0, stream>>>(rnk, part, G);
    k_winner   <<<(N + TB - 1) / TB, TB, 0, stream>>>(pts, bidx, rnk, last_pos, N);
    const int total = G * PMAX;                     // 16.7M lanes, one b128 store each
    k_write    <<<(total + TB - 1) / TB, TB, 0, stream>>>(pts, last_pos, out, G);
    k_zero_tail<<<(G + TB - 1) / TB, TB, 0, stream>>>(last_pos, G);
}